// DeformableConv2d_35545149342350
// MI455X (gfx1250) — compile-verified
//
#include <hip/hip_runtime.h>
#include <hip/hip_bf16.h>

typedef __attribute__((ext_vector_type(2))) float v2f;
typedef __attribute__((ext_vector_type(8))) float v8f;

#define Hh 128
#define Ww 128
#define CIN 64
#define COUT 128
#define HW 16384           // 128*128 = 1<<14
#define BSTRIDE 66         // pixel-major B-tile row stride (even => 8B-aligned pairs)

// ---------------------------------------------------------------------------
// Kernel 0: repack main-conv weights  wconv[o][c][m] -> wrep[m][o][c]
// so the WMMA A fragment (o fixed per lane, c0/c0+1 adjacent) is one b64 load.
// ---------------------------------------------------------------------------
__global__ void __launch_bounds__(256)
repack_w_kernel(const float* __restrict__ wconv, float* __restrict__ wrep)
{
    int i = blockIdx.x * 256 + threadIdx.x;        // over 128*64*9 = 73728
    if (i < COUT * CIN * 9) {
        int m = i % 9;
        int c = (i / 9) % CIN;
        int o = i / (9 * CIN);
        wrep[(m * COUT + o) * CIN + c] = wconv[i];
    }
}

// ---------------------------------------------------------------------------
// Kernel 1: offset-predicting conv (18 = 2*3*3 output channels), pad=1.
// 16x16 spatial tile per block; x tiled through LDS 4 channels at a time.
// ---------------------------------------------------------------------------
__global__ void __launch_bounds__(256)
offset_conv_kernel(const float* __restrict__ x,
                   const float* __restrict__ w_off,
                   const float* __restrict__ b_off,
                   float* __restrict__ off)
{
    const int n  = blockIdx.z;
    const int bh = blockIdx.y;
    const int bw = blockIdx.x;
    const int ty = threadIdx.y, tx = threadIdx.x;
    const int h  = bh * 16 + ty, w = bw * 16 + tx;
    const int tid = ty * 16 + tx;

    __shared__ float tile[4][18][19];

    float acc[18];
#pragma unroll
    for (int ch = 0; ch < 18; ++ch) acc[ch] = b_off[ch];

    for (int c0 = 0; c0 < CIN; c0 += 4) {
        __syncthreads();
        for (int e = tid; e < 4 * 18 * 18; e += 256) {
            int cc = e / 324, r = e % 324;
            int ih = bh * 16 - 1 + r / 18;
            int iw = bw * 16 - 1 + r % 18;
            float v = 0.0f;
            if (ih >= 0 && ih < Hh && iw >= 0 && iw < Ww)
                v = x[((n * CIN + c0 + cc) << 14) + ih * Ww + iw];
            tile[cc][r / 18][r % 18] = v;
        }
        __syncthreads();
#pragma unroll
        for (int cc = 0; cc < 4; ++cc) {
            float xv[9];
#pragma unroll
            for (int i = 0; i < 3; ++i)
#pragma unroll
                for (int j = 0; j < 3; ++j)
                    xv[i * 3 + j] = tile[cc][ty + i][tx + j];
#pragma unroll
            for (int ch = 0; ch < 18; ++ch) {
                const float* wp = w_off + ((ch * CIN + c0 + cc) * 9);
                float s = 0.0f;
#pragma unroll
                for (int t = 0; t < 9; ++t) s = fmaf(xv[t], wp[t], s);
                acc[ch] += s;
            }
        }
    }
#pragma unroll
    for (int ch = 0; ch < 18; ++ch)
        off[((n * 18 + ch) << 14) + h * Ww + w] = acc[ch];
}

// ---------------------------------------------------------------------------
// Kernel 2: fused bilinear deform-sampling + im2col GEMM via fp32 WMMA.
// Block: 256 threads (8 waves). blockIdx.x = 64-pixel tile, blockIdx.y = n.
// Wave wv owns Cout rows [16wv,16wv+16); 4 N-tiles of 16 pixels; K = 64*9.
// Double-buffered B tile in LDS, one barrier per tap; gathers for tap m+1
// are issued before the WMMA block of tap m to hide their latency.
// ---------------------------------------------------------------------------
__global__ void __launch_bounds__(256)
deform_gemm_kernel(const float* __restrict__ x,
                   const float* __restrict__ off,
                   const float* __restrict__ wrep,
                   const float* __restrict__ bconv,
                   float* __restrict__ out)
{
    const int n     = blockIdx.y;
    const int pix0  = blockIdx.x * 64;
    const int tid   = threadIdx.x;
    const int lane  = tid & 31;
    const int wv    = tid >> 5;
    const int obase = wv * 16;
    const int l15   = lane & 15;
    const int hiH   = lane >> 4;          // 0: lanes 0-15, 1: lanes 16-31
    const int pix   = tid & 63;           // this thread's fixed pixel
    const int cbase = tid >> 6;           // 0..3; c = cbase + 4r, r=0..15

    __shared__ int   sIx[576 * 4];        // gather indices per (tap,pix)
    __shared__ float sWg[576 * 4];        // bilinear weights per (tap,pix)
    __shared__ float sB[2][64 * BSTRIDE]; // pixel-major B tiles (double buffer)

    // ---- Phase 0: per-(tap,pixel) sampling tables (reference semantics:
    //      p = offset + tap base, no (h,w) term; clip, then weights from
    //      the clipped coordinates).
    for (int pt = tid; pt < 576; pt += 256) {
        int m = pt >> 6, p = pt & 63;
        int pg = pix0 + p;
        float ph = off[((n * 18 + 2 * m    ) << 14) + pg] + (float)(m / 3);
        float pw = off[((n * 18 + 2 * m + 1) << 14) + pg] + (float)(m % 3);
        float fh = floorf(ph), fw = floorf(pw);
        int hf = min(max((int)fh, 0), Hh - 1);
        int wf = min(max((int)fw, 0), Ww - 1);
        int hc = min(max((int)fh + 1, 0), Hh - 1);
        int wc = min(max((int)fw + 1, 0), Ww - 1);
        float hff = (float)hf, wff = (float)wf;
        float hcf = (float)hc, wcf = (float)wc;
        sIx[pt * 4 + 0] = hf * Ww + wf;  sWg[pt * 4 + 0] = (hcf - ph) * (wcf - pw);
        sIx[pt * 4 + 1] = hf * Ww + wc;  sWg[pt * 4 + 1] = (hcf - ph) * (pw - wff);
        sIx[pt * 4 + 2] = hc * Ww + wf;  sWg[pt * 4 + 2] = (ph - hff) * (wcf - pw);
        sIx[pt * 4 + 3] = hc * Ww + wc;  sWg[pt * 4 + 3] = (ph - hff) * (pw - wff);
    }
    __syncthreads();

    v8f acc[4] = {};
    const float* xcb = x + ((size_t)(n * CIN + cbase) << 14);  // this thread's c-base plane

    // ---- prologue: sample tap 0 into buf 0
    {
        int pt = pix * 4;                       // tap 0 table entry
        float w0 = sWg[pt + 0], w1 = sWg[pt + 1], w2 = sWg[pt + 2], w3 = sWg[pt + 3];
        const float* p0 = xcb + sIx[pt + 0];
        const float* p1 = xcb + sIx[pt + 1];
        const float* p2 = xcb + sIx[pt + 2];
        const float* p3 = xcb + sIx[pt + 3];
        float* sp = &sB[0][pix * BSTRIDE + cbase];
#pragma unroll
        for (int r = 0; r < 16; ++r) {
            int d = r << 16;                    // 4 planes * 16384 per c-step
            sp[4 * r] = w0 * p0[d] + w1 * p1[d] + w2 * p2[d] + w3 * p3[d];
        }
    }
    __syncthreads();

    for (int m = 0; m < 9; ++m) {
        // ---- (a) issue next tap's gathers first (latency hides under WMMA)
        float vals[16];
        if (m < 8) {
            int pt = (((m + 1) << 6) + pix) * 4;
            float w0 = sWg[pt + 0], w1 = sWg[pt + 1], w2 = sWg[pt + 2], w3 = sWg[pt + 3];
            const float* p0 = xcb + sIx[pt + 0];
            const float* p1 = xcb + sIx[pt + 1];
            const float* p2 = xcb + sIx[pt + 2];
            const float* p3 = xcb + sIx[pt + 3];
#pragma unroll
            for (int r = 0; r < 16; ++r) {
                int d = r << 16;
                vals[r] = w0 * p0[d] + w1 * p1[d] + w2 * p2[d] + w3 * p3[d];
            }
        }

        // ---- (b) WMMA block for tap m: 16 K-steps x 4 N-tiles.
        // A 16x4 (ISA 7.12.2): VGPR0 = K0|K2 by lane half, VGPR1 = K1|K3.
        // A pair (o, c0),(o, c0+1) is contiguous in wrep -> one b64 load.
        // B pair (c0,pix),(c0+1,pix) is contiguous in pixel-major sB -> ds_load_b64.
        const float* ab = wrep + (((m * COUT + obase + l15) << 6) + 2 * hiH);
        const float* bb = &sB[m & 1][2 * hiH + l15 * BSTRIDE];
#pragma unroll 4
        for (int kk = 0; kk < 16; ++kk) {
            v2f a = *(const v2f*)(ab + (kk << 2));
#pragma unroll
            for (int j = 0; j < 4; ++j) {
                v2f b = *(const v2f*)(bb + j * 16 * BSTRIDE + (kk << 2));
                acc[j] = __builtin_amdgcn_wmma_f32_16x16x4_f32(
                    false, a, false, b, (short)0, acc[j], false, false);
            }
        }

        // ---- (c) commit next tap's tile, then the single per-tap barrier
        if (m < 8) {
            float* sp = &sB[(m + 1) & 1][pix * BSTRIDE + cbase];
#pragma unroll
            for (int r = 0; r < 16; ++r) sp[4 * r] = vals[r];
        }
        __syncthreads();
    }

    // ---- epilogue: C layout = VGPR i -> M=i (lanes 0-15) / M=i+8 (lanes 16-31)
    const int hi = hiH ? 8 : 0;
#pragma unroll
    for (int j = 0; j < 4; ++j) {
        int pg = pix0 + j * 16 + l15;
#pragma unroll
        for (int i = 0; i < 8; ++i) {
            int o = obase + i + hi;
            out[((size_t)(n * COUT + o) << 14) + pg] = acc[j][i] + bconv[o];
        }
    }
}

// ---------------------------------------------------------------------------
extern "C" void kernel_launch(void* const* d_in, const int* in_sizes, int n_in,
                              void* d_out, int out_size, void* d_ws, size_t ws_size,
                              hipStream_t stream)
{
    const float* x      = (const float*)d_in[0];
    const float* w_off  = (const float*)d_in[1];
    const float* b_off  = (const float*)d_in[2];
    const float* w_conv = (const float*)d_in[3];
    const float* b_conv = (const float*)d_in[4];
    float* out  = (float*)d_out;
    float* off  = (float*)d_ws;                 // 8*18*128*128 floats = 9.44 MB
    float* wrep = off + 8 * 18 * HW;            // + 73728 floats = 295 KB

    repack_w_kernel<<<288, 256, 0, stream>>>(w_conv, wrep);

    dim3 g1(8, 8, 8), b1(16, 16);
    offset_conv_kernel<<<g1, b1, 0, stream>>>(x, w_off, b_off, off);

    dim3 g2(256, 8), b2(256);                   // 2048 workgroups, 8 waves each
    deform_gemm_kernel<<<g2, b2, 0, stream>>>(x, off, wrep, b_conv, out);
}